// GraphConv_6734508720141
// MI455X (gfx1250) — compile-verified
//
#include <hip/hip_runtime.h>
#include <hip/hip_bf16.h>

// ---------------------------------------------------------------------------
// GraphConv: out = segment_sum_rows( adj_vals * (x@W)[adj_cols] ) + bias
//   Stage 0: out[n][f] = bias[f]
//   Stage 1: support = x @ W   (WMMA f32 16x16x4, W pair-swizzled in LDS)
//   Stage 2: edge scatter with f32 atomics (memory/atomic bound, ~70us floor)
// ---------------------------------------------------------------------------

typedef float v2f __attribute__((ext_vector_type(2)));
typedef float v4f __attribute__((ext_vector_type(4)));
typedef float v8f __attribute__((ext_vector_type(8)));

#define FDIM 128   // F_IN == F_OUT == 128

// ---------------------------------------------------------------------------
// Stage 0: broadcast bias into every output row.
// ---------------------------------------------------------------------------
__global__ __launch_bounds__(256) void gc_init_bias(float* __restrict__ out,
                                                    const float* __restrict__ bias,
                                                    int total) {
  int i = blockIdx.x * 256 + threadIdx.x;
  if (i < total) out[i] = bias[i & (FDIM - 1)];
}

// ---------------------------------------------------------------------------
// Stage 1: support = x @ W via V_WMMA_F32_16X16X4_F32.
// Block = 256 threads = 8 waves. Block stages W (64KB) into LDS in a
// K-pair-swizzled layout: lds_w[kk*FDIM + col] = { W[2kk][col], W[2kk+1][col] }
// so each lane's B fragment is ONE aligned ds_load_b64 into an even VGPR
// pair (no v_mov shuffles between DS and WMMA).
//
// Fragment layouts (ISA 7.12.2, wave32):
//   A 16x4 f32 : lanes 0-15 carry M=lane, {K0,K1}; lanes 16-31 {K2,K3}.
//   B 4x16 f32 : VGPR0 = row K0 (lanes 0-15) / K2 (lanes 16-31), VGPR1 = K1/K3.
//   C/D 16x16  : VGPR r -> M=r (lanes 0-15) / M=r+8 (lanes 16-31), N=lane%16.
// Each wave owns a 16x128 stripe: 8 accumulator tiles, 32 wmma steps each.
// ---------------------------------------------------------------------------
__global__ __launch_bounds__(256) void gc_gemm_wmma(const float* __restrict__ x,
                                                    const float* __restrict__ W,
                                                    float* __restrict__ support,
                                                    int n_nodes) {
  __shared__ v2f lds_w[(FDIM / 2) * FDIM];   // 64 KB of 320 KB/WGP

  // Cooperative pair-swizzled stage of W into LDS.
  // pair p -> kk = p/128 (K pair index), col = p%128; consecutive lanes hit
  // consecutive columns of the same two W rows -> coalesced 128B segments.
  for (int p = threadIdx.x; p < (FDIM / 2) * FDIM; p += 256) {
    const int kk  = p >> 7;        // 0..63
    const int col = p & (FDIM - 1);
    v2f pr;
    pr.x = W[(2 * kk) * FDIM + col];
    pr.y = W[(2 * kk + 1) * FDIM + col];
    lds_w[p] = pr;
  }
  __syncthreads();

  const int wave  = threadIdx.x >> 5;      // 0..7 (wave32)
  const int lane  = threadIdx.x & 31;
  const int mtile = blockIdx.x * 8 + wave; // 16 rows per tile
  if (mtile * 16 >= n_nodes) return;       // wave-uniform: EXEC all-1s below

  const int lhalf = lane & 15;
  const int kp    = (lane >= 16) ? 1 : 0;  // K-pair sub-offset for half-wave
  const size_t arow = (size_t)(mtile * 16 + lhalf) * FDIM;

  v8f acc[8] = {};                         // 8 N-tiles x 8 VGPRs = 16x128 out

  for (int k = 0; k < FDIM; k += 4) {
    // A fragment: two consecutive K values -> one 8B global load.
    v2f a = *(const v2f*)(x + arow + k + 2 * kp);
    // B fragments: one ds_load_b64 each; 8 tiles share one base address
    // with immediate offsets t*128 bytes.
    const v2f* __restrict__ bbase = &lds_w[(k / 2 + kp) * FDIM + lhalf];
#pragma unroll
    for (int t = 0; t < 8; ++t) {
      const v2f b = bbase[t * 16];
      acc[t] = __builtin_amdgcn_wmma_f32_16x16x4_f32(
          /*neg_a=*/false, a, /*neg_b=*/false, b,
          /*c_mod=*/(short)0, acc[t], /*reuse_a=*/false, /*reuse_b=*/false);
    }
  }

  // Write back D tiles.
  const size_t mbase = (size_t)(mtile * 16 + ((lane >= 16) ? 8 : 0));
#pragma unroll
  for (int t = 0; t < 8; ++t) {
#pragma unroll
    for (int r = 0; r < 8; ++r) {
      support[(mbase + r) * FDIM + t * 16 + lhalf] = acc[t][r];
    }
  }
}

// ---------------------------------------------------------------------------
// Stage 2: for each edge e: out[row[e]] += val[e] * support[col[e]].
// One wave per edge; indices are wave-uniform (scalar loads); each lane
// handles 4 features -> one global_load_b128 + 4 global_atomic_add_f32.
// out fits in L2 (25.6MB of 192MB) so atomics resolve on-chip.
// ---------------------------------------------------------------------------
__global__ __launch_bounds__(256) void gc_spmm_scatter(const float* __restrict__ vals,
                                                       const int* __restrict__ rows,
                                                       const int* __restrict__ cols,
                                                       const float* __restrict__ support,
                                                       float* __restrict__ out,
                                                       int n_edges) {
  const int wave = threadIdx.x >> 5;
  const int lane = threadIdx.x & 31;
  const int e = blockIdx.x * 8 + wave;
  if (e >= n_edges) return;

  const int r = rows[e];
  const int c = cols[e];
  const float v = vals[e];

  const v4f s = *(const v4f*)(support + (size_t)c * FDIM + lane * 4);
  float* dst = out + (size_t)r * FDIM + lane * 4;
  atomicAdd(dst + 0, v * s.x);
  atomicAdd(dst + 1, v * s.y);
  atomicAdd(dst + 2, v * s.z);
  atomicAdd(dst + 3, v * s.w);
}

// ---------------------------------------------------------------------------
// Launch
// Inputs (setup_inputs order): x, W, bias, adj_vals, adj_rows, adj_cols
// ---------------------------------------------------------------------------
extern "C" void kernel_launch(void* const* d_in, const int* in_sizes, int n_in,
                              void* d_out, int out_size, void* d_ws, size_t ws_size,
                              hipStream_t stream) {
  const float* x    = (const float*)d_in[0];
  const float* W    = (const float*)d_in[1];
  const float* bias = (const float*)d_in[2];
  const float* vals = (const float*)d_in[3];
  const int*   rows = (const int*)d_in[4];
  const int*   cols = (const int*)d_in[5];
  float* out = (float*)d_out;

  const int n_nodes = in_sizes[0] / FDIM;
  const int n_edges = in_sizes[3];

  float* support = (float*)d_ws;  // n_nodes * FDIM floats (25.6 MB)

  // Stage 0: out = bias (broadcast per row)
  {
    const int total = n_nodes * FDIM;
    gc_init_bias<<<(total + 255) / 256, 256, 0, stream>>>(out, bias, total);
  }

  // Stage 1: support = x @ W  (8 waves/block, 16 rows per wave)
  {
    const int mtiles = (n_nodes + 15) / 16;
    const int blocks = (mtiles + 7) / 8;
    gc_gemm_wmma<<<blocks, 256, 0, stream>>>(x, W, support, n_nodes);
  }

  // Stage 2: edge scatter (1 wave per edge, 8 edges per block)
  {
    const int blocks = (n_edges + 7) / 8;
    gc_spmm_scatter<<<blocks, 256, 0, stream>>>(vals, rows, cols, support, out, n_edges);
  }
}